// TransformerBlock_1984274891513
// MI455X (gfx1250) — compile-verified
//
#include <hip/hip_runtime.h>

// ---------------------------------------------------------------------------
// PointTransformerConv block for MI455X (gfx1250, wave32, WMMA).
// fp32 WMMA (V_WMMA_F32_16X16X4_F32); 3-pass edge recompute keeps all node
// tensors (~154MB) L2-resident and avoids 435MB+ edge intermediates.
// Weights live in LDS k-pair interleaved so every WMMA B-fragment is a single
// conflict-free ds_load_b64 into an even VGPR pair (no v_mov glue).
// ---------------------------------------------------------------------------

#define AP 68    // LDS pitch (floats) for 16x64 activation tiles (conflict-free)
#define P2 160   // LDS pitch (floats) per k-PAIR row of interleaved weights:
                 // banks of row r+1 are offset by 32 -> half-waves disjoint

typedef __attribute__((ext_vector_type(2))) float v2f;
typedef __attribute__((ext_vector_type(8))) float v8f;

__device__ __forceinline__ v8f wmma4(v2f a, v2f b, v8f c) {
  // D(16x16,f32) = A(16x4,f32) x B(4x16,f32) + C
  return __builtin_amdgcn_wmma_f32_16x16x4_f32(false, a, false, b, (short)0, c,
                                               false, false);
}

// Intra-wave LDS producer->consumer ordering (CDNA5 split counters).
__device__ __forceinline__ void wave_lds_sync() {
  asm volatile("s_wait_dscnt 0" ::: "memory");
}

// Monotonic uint key for float max-reduction via atomicMax(u32).
__device__ __forceinline__ unsigned fmax_key(float x) {
  unsigned u = __float_as_uint(x);
  return (u & 0x80000000u) ? ~u : (u | 0x80000000u);
}
__device__ __forceinline__ float fmax_unkey(unsigned k) {
  return (k & 0x80000000u) ? __uint_as_float(k & 0x7fffffffu)
                           : __uint_as_float(~k);
}

// 16x64 @ 64x64 GEMM. A rows from LDS activation tile (contiguous k-pairs ->
// ds_load_b64), B from LDS interleaved weights (one ds_load_b64 per fragment).
__device__ __forceinline__ void gemm_64(const float* __restrict__ actA,
                                        const float* __restrict__ wt2,
                                        int col, int kof, v8f acc[4]) {
  const v2f* __restrict__ arow = (const v2f*)(actA + col * AP);  // col*AP even
  const v2f* __restrict__ wrow = (const v2f*)wt2;
#pragma unroll 4
  for (int s = 0; s < 16; ++s) {
    const int k0 = 4 * s + kof;                 // even
    const v2f af = arow[k0 >> 1];
    const int rb = (k0 >> 1) * (P2 / 2);        // v2f units
#pragma unroll
    for (int t = 0; t < 4; ++t) {
      const v2f bf = wrow[rb + 16 * t + col];
      acc[t] = wmma4(af, bf, acc[t]);
    }
  }
}

// ---------------------------------------------------------------------------
__global__ void init_kernel(unsigned* __restrict__ amax_u,
                            float* __restrict__ esum,
                            unsigned* __restrict__ outkey, size_t n) {
  size_t i = (size_t)blockIdx.x * blockDim.x + threadIdx.x;
  if (i < n) {
    amax_u[i] = 0u;   // 0.0f bits; attn logits are post-ReLU >= 0
    esum[i] = 0.0f;
    outkey[i] = 0u;   // minimum possible key (never survives: self-loops)
  }
}

// ---------------------------------------------------------------------------
// Node phase: xr = relu(x @ W_in^T + b_in); alpha_src/dst = xr @ W_{src,dst}^T;
// xv = xr @ W_lin^T.  Fused per 16-node tile, 4 WMMA GEMMs.
__launch_bounds__(128)
__global__ void node_proj_kernel(const float* __restrict__ x,
                                 const float* __restrict__ W_in,
                                 const float* __restrict__ b_in,
                                 const float* __restrict__ W_src,
                                 const float* __restrict__ W_dst,
                                 const float* __restrict__ W_lin,
                                 float* __restrict__ a_src,
                                 float* __restrict__ a_dst,
                                 float* __restrict__ xv, int N) {
  extern __shared__ float lds[];
  float* wi = lds;                 // 32 row-pairs x P2
  float* ws_ = wi + 32 * P2;
  float* wd = ws_ + 32 * P2;
  float* wl = wd + 32 * P2;
  float* actbase = wl + 32 * P2;   // 4 waves x (16 x AP)

  // Stage transposed + k-pair interleaved: wt2[(k>>1)*P2 + 2*o + (k&1)] = W[o][k]
  for (int i = threadIdx.x; i < 64 * 64; i += blockDim.x) {
    int o = i >> 6, k = i & 63;
    int di = (k >> 1) * P2 + 2 * o + (k & 1);
    wi[di] = W_in[i];
    ws_[di] = W_src[i];
    wd[di] = W_dst[i];
    wl[di] = W_lin[i];
  }
  __syncthreads();

  const int lane = threadIdx.x & 31;
  const int wid = threadIdx.x >> 5;
  const int col = lane & 15;
  const int half = lane >> 4;
  const int kof = half << 1;  // 0 or 2

  float* act = actbase + wid * (16 * AP);

  float binr[4];
#pragma unroll
  for (int t = 0; t < 4; ++t) binr[t] = b_in[16 * t + col];

  const int ntiles = (N + 15) >> 4;
  const int gw = blockIdx.x * 4 + wid;
  const int nw = gridDim.x * 4;

  for (int tile = gw; tile < ntiles; tile += nw) {
    int row = tile * 16 + col;
    if (row >= N) row = N - 1;
    const v2f* xr = (const v2f*)(x + (size_t)row * 64);

    // layer: xr_relu = relu(x @ W_in^T + b_in)
    v8f h[4] = {};
    {
      const v2f* wrow = (const v2f*)wi;
#pragma unroll 4
      for (int s = 0; s < 16; ++s) {
        const int k0 = 4 * s + kof;
        const v2f af = xr[k0 >> 1];
        const int rb = (k0 >> 1) * (P2 / 2);
#pragma unroll
        for (int t = 0; t < 4; ++t)
          h[t] = wmma4(af, wrow[rb + 16 * t + col], h[t]);
      }
    }
#pragma unroll
    for (int t = 0; t < 4; ++t)
#pragma unroll
      for (int r = 0; r < 8; ++r) {
        float v = h[t][r] + binr[t];
        act[(r + 8 * half) * AP + 16 * t + col] = v > 0.f ? v : 0.f;
      }
    wave_lds_sync();

    v8f as4[4] = {}, ad4[4] = {}, xv4[4] = {};
    gemm_64(act, ws_, col, kof, as4);
    gemm_64(act, wd, col, kof, ad4);
    gemm_64(act, wl, col, kof, xv4);

    const int nb = tile * 16;
#pragma unroll
    for (int t = 0; t < 4; ++t)
#pragma unroll
      for (int r = 0; r < 8; ++r) {
        int nrow = nb + r + 8 * half;
        if (nrow >= N) nrow = N - 1;
        size_t off = (size_t)nrow * 64 + 16 * t + col;
        a_src[off] = as4[t][r];
        a_dst[off] = ad4[t][r];
        xv[off] = xv4[t][r];
      }
    wave_lds_sync();  // done with act before next tile overwrites
  }
}

// ---------------------------------------------------------------------------
// Edge phase (recomputed 3x):
//  PASS 1: a = attn MLP; segment-max into amax (raw-bit atomicMax, a>=0).
//  PASS 2: e = exp(a - amax[dst]); atomicAdd into esum.
//  PASS 3: w = e/esum[dst]; msg = w*(xv[src]+delta); keyed atomicMax into out.
template <int PASS>
__launch_bounds__(128)
__global__ void edge_pass_kernel(
    const float* __restrict__ pos, const int* __restrict__ ei,
    const float* __restrict__ pW1, const float* __restrict__ pb1,
    const float* __restrict__ pW2, const float* __restrict__ pb2,
    const float* __restrict__ aW1, const float* __restrict__ ab1,
    const float* __restrict__ aW2, const float* __restrict__ ab2,
    const float* __restrict__ a_src, const float* __restrict__ a_dst,
    const float* __restrict__ xv, float* __restrict__ amax,
    float* __restrict__ esum, unsigned* __restrict__ outkey, int E, int N) {
  extern __shared__ float lds[];
  float* wp1 = lds;               // 2 row-pairs x P2 (K padded 3->4, row 3 = 0)
  float* wp2 = wp1 + 2 * P2;      // 32 row-pairs x P2
  float* wa1 = wp2 + 32 * P2;
  float* wa2 = wa1 + 32 * P2;
  float* actbase = wa2 + 32 * P2;                    // 4 waves x 2 x (16 x AP)
  int* idxbase = (int*)(actbase + 4 * 2 * 16 * AP);  // 4 waves x 32 ints

  // pW2/aW1/aW2 are already [k][o] in memory; interleave k-pairs into LDS.
  for (int i = threadIdx.x; i < 64 * 64; i += blockDim.x) {
    int k = i >> 6, o = i & 63;
    int di = (k >> 1) * P2 + 2 * o + (k & 1);
    wp2[di] = pW2[i];
    wa1[di] = aW1[i];
    wa2[di] = aW2[i];
  }
  for (int i = threadIdx.x; i < 4 * 64; i += blockDim.x) {
    int k = i >> 6, o = i & 63;
    wp1[(k >> 1) * P2 + 2 * o + (k & 1)] = (k < 3) ? pW1[k * 64 + o] : 0.f;
  }
  __syncthreads();

  const int lane = threadIdx.x & 31;
  const int wid = threadIdx.x >> 5;
  const int col = lane & 15;
  const int half = lane >> 4;
  const int kof = half << 1;

  float* act1 = actbase + wid * (2 * 16 * AP);  // holds h1, then delta
  float* act2 = act1 + 16 * AP;                 // holds h2
  int* sidx = idxbase + wid * 32;
  int* didx = sidx + 16;

  float pb1r[4], pb2r[4], ab1r[4], ab2r[4];
#pragma unroll
  for (int t = 0; t < 4; ++t) {
    pb1r[t] = pb1[16 * t + col];
    pb2r[t] = pb2[16 * t + col];
    ab1r[t] = ab1[16 * t + col];
    ab2r[t] = ab2[16 * t + col];
  }

  const int EE = E + N;  // edges + self loops (multiple of 16 at these sizes)
  const int ntiles = (EE + 15) >> 4;
  const int gw = blockIdx.x * 4 + wid;
  const int nw = gridDim.x * 4;

  for (int tile = gw; tile < ntiles; tile += nw) {
    int e = tile * 16 + col;
    if (e >= EE) e = EE - 1;  // safety net (no tail at these sizes)
    int s, d;
    if (e < E) {
      s = ei[e];
      d = ei[E + e];
    } else {
      s = d = e - E;  // self loop
    }
    if (lane < 16) {
      sidx[col] = s;
      didx[col] = d;
    }
    wave_lds_sync();

    // ---- pos_nn layer 1: h1 = relu(pd @ pW1 + pb1), K padded to 4
    float pd0 = pos[(size_t)d * 3 + kof] - pos[(size_t)s * 3 + kof];  // k in {0,2}
    float pd1 = (kof + 1 < 3)
                    ? (pos[(size_t)d * 3 + kof + 1] - pos[(size_t)s * 3 + kof + 1])
                    : 0.f;
    v2f af;
    af.x = pd0;
    af.y = pd1;
    v8f h[4] = {};
    {
      const v2f* wrow = (const v2f*)wp1;
      const int rb = (kof >> 1) * (P2 / 2);
#pragma unroll
      for (int t = 0; t < 4; ++t)
        h[t] = wmma4(af, wrow[rb + 16 * t + col], h[t]);
    }
#pragma unroll
    for (int t = 0; t < 4; ++t)
#pragma unroll
      for (int r = 0; r < 8; ++r) {
        float v = h[t][r] + pb1r[t];
        act1[(r + 8 * half) * AP + 16 * t + col] = v > 0.f ? v : 0.f;
      }
    wave_lds_sync();

    // ---- pos_nn layer 2: delta = relu(h1 @ pW2 + pb2)   (overwrites act1)
    v8f dl[4] = {};
    gemm_64(act1, wp2, col, kof, dl);
    wave_lds_sync();  // all h1 reads complete before overwrite
#pragma unroll
    for (int t = 0; t < 4; ++t)
#pragma unroll
      for (int r = 0; r < 8; ++r) {
        float v = dl[t][r] + pb2r[t];
        act1[(r + 8 * half) * AP + 16 * t + col] = v > 0.f ? v : 0.f;
      }
    wave_lds_sync();

    // ---- attn layer 1: h2 = relu((a_dst[d]-a_src[s]+delta) @ aW1 + ab1)
    const v2f* prow_d = (const v2f*)(a_dst + (size_t)d * 64);
    const v2f* prow_s = (const v2f*)(a_src + (size_t)s * 64);
    const v2f* drow = (const v2f*)(act1 + col * AP);
    v8f g[4] = {};
    {
      const v2f* wrow = (const v2f*)wa1;
#pragma unroll 4
      for (int st = 0; st < 16; ++st) {
        const int k0 = 4 * st + kof;
        const v2f pdv = prow_d[k0 >> 1];
        const v2f psv = prow_s[k0 >> 1];
        const v2f dlv = drow[k0 >> 1];
        v2f af2;
        af2.x = pdv.x - psv.x + dlv.x;
        af2.y = pdv.y - psv.y + dlv.y;
        const int rb = (k0 >> 1) * (P2 / 2);
#pragma unroll
        for (int t = 0; t < 4; ++t)
          g[t] = wmma4(af2, wrow[rb + 16 * t + col], g[t]);
      }
    }
#pragma unroll
    for (int t = 0; t < 4; ++t)
#pragma unroll
      for (int r = 0; r < 8; ++r) {
        float v = g[t][r] + ab1r[t];
        act2[(r + 8 * half) * AP + 16 * t + col] = v > 0.f ? v : 0.f;
      }
    wave_lds_sync();

    // ---- attn layer 2: a = relu(h2 @ aW2 + ab2)
    v8f av[4] = {};
    gemm_64(act2, wa2, col, kof, av);

    // ---- pass-specific epilogue (elements in C/D layout: M=r8, ch=c)
#pragma unroll
    for (int t = 0; t < 4; ++t) {
      const int c = 16 * t + col;
#pragma unroll
      for (int r = 0; r < 8; ++r) {
        const int r8 = r + 8 * half;
        float a_el = av[t][r] + ab2r[t];
        a_el = a_el > 0.f ? a_el : 0.f;
        const int dn = didx[r8];
        const size_t off = (size_t)dn * 64 + c;
        if (PASS == 1) {
          atomicMax((unsigned*)amax + off, __float_as_uint(a_el));  // a_el >= 0
        } else {
          const float ex = __expf(a_el - amax[off]);
          if (PASS == 2) {
            atomicAdd(esum + off, ex);
          } else {
            const float w = ex / esum[off];
            const int sn = sidx[r8];
            const float val =
                w * (xv[(size_t)sn * 64 + c] + act1[r8 * AP + c]);  // delta
            atomicMax(outkey + off, fmax_key(val));
          }
        }
      }
    }
    wave_lds_sync();  // sidx/didx/act1 reads done before next tile overwrites
  }
}

// ---------------------------------------------------------------------------
// Output phase: y = relu(decode(outkey) @ W_out^T + b_out)
__launch_bounds__(128)
__global__ void out_kernel(const unsigned* __restrict__ outkey,
                           const float* __restrict__ W_out,
                           const float* __restrict__ b_out,
                           float* __restrict__ y, int N) {
  extern __shared__ float lds[];
  float* wo = lds;  // 32 row-pairs x P2
  for (int i = threadIdx.x; i < 64 * 64; i += blockDim.x) {
    int o = i >> 6, k = i & 63;
    wo[(k >> 1) * P2 + 2 * o + (k & 1)] = W_out[i];  // [out][in] -> interleaved
  }
  __syncthreads();

  const int lane = threadIdx.x & 31;
  const int wid = threadIdx.x >> 5;
  const int col = lane & 15;
  const int half = lane >> 4;
  const int kof = half << 1;

  float bor[4];
#pragma unroll
  for (int t = 0; t < 4; ++t) bor[t] = b_out[16 * t + col];

  const int ntiles = (N + 15) >> 4;
  const int gw = blockIdx.x * 4 + wid;
  const int nw = gridDim.x * 4;

  for (int tile = gw; tile < ntiles; tile += nw) {
    int row = tile * 16 + col;
    if (row >= N) row = N - 1;
    const unsigned* kr = outkey + (size_t)row * 64;

    v8f h[4] = {};
    {
      const v2f* wrow = (const v2f*)wo;
#pragma unroll 4
      for (int s = 0; s < 16; ++s) {
        const int k0 = 4 * s + kof;
        v2f af;
        af.x = fmax_unkey(kr[k0]);
        af.y = fmax_unkey(kr[k0 + 1]);
        const int rb = (k0 >> 1) * (P2 / 2);
#pragma unroll
        for (int t = 0; t < 4; ++t)
          h[t] = wmma4(af, wrow[rb + 16 * t + col], h[t]);
      }
    }
    const int nb = tile * 16;
#pragma unroll
    for (int t = 0; t < 4; ++t)
#pragma unroll
      for (int r = 0; r < 8; ++r) {
        int nrow = nb + r + 8 * half;
        if (nrow >= N) nrow = N - 1;
        float v = h[t][r] + bor[t];
        y[(size_t)nrow * 64 + 16 * t + col] = v > 0.f ? v : 0.f;
      }
  }
}

// ---------------------------------------------------------------------------
extern "C" void kernel_launch(void* const* d_in, const int* in_sizes, int n_in,
                              void* d_out, int out_size, void* d_ws,
                              size_t ws_size, hipStream_t stream) {
  (void)n_in;
  (void)out_size;
  (void)ws_size;
  const float* x = (const float*)d_in[0];
  const float* pos = (const float*)d_in[1];
  const int* ei = (const int*)d_in[2];
  const float* W_in = (const float*)d_in[3];
  const float* b_in = (const float*)d_in[4];
  const float* W_out = (const float*)d_in[5];
  const float* b_out = (const float*)d_in[6];
  const float* W_lin = (const float*)d_in[7];
  const float* W_src = (const float*)d_in[8];
  const float* W_dst = (const float*)d_in[9];
  const float* pW1 = (const float*)d_in[10];
  const float* pb1 = (const float*)d_in[11];
  const float* pW2 = (const float*)d_in[12];
  const float* pb2 = (const float*)d_in[13];
  const float* aW1 = (const float*)d_in[14];
  const float* ab1 = (const float*)d_in[15];
  const float* aW2 = (const float*)d_in[16];
  const float* ab2 = (const float*)d_in[17];

  const int N = in_sizes[0] / 64;
  const int E = in_sizes[2] / 2;
  const size_t NC = (size_t)N * 64;

  // Workspace layout (fp32 node tensors; all fit in MI455X's 192MB L2)
  float* ws = (float*)d_ws;
  float* a_src = ws;
  float* a_dst = ws + NC;
  float* xv = ws + 2 * NC;
  float* amax = ws + 3 * NC;
  float* esum = ws + 4 * NC;
  unsigned* outkey = (unsigned*)(ws + 5 * NC);

  {
    const int bs = 256;
    const int gb = (int)((NC + bs - 1) / bs);
    init_kernel<<<gb, bs, 0, stream>>>((unsigned*)amax, esum, outkey, NC);
  }

  const size_t node_lds = (size_t)(4 * 32 * P2 + 4 * 16 * AP) * sizeof(float);
  node_proj_kernel<<<dim3(512), dim3(128), node_lds, stream>>>(
      x, W_in, b_in, W_src, W_dst, W_lin, a_src, a_dst, xv, N);

  const size_t edge_lds =
      (size_t)(2 * P2 + 3 * 32 * P2 + 4 * 2 * 16 * AP) * sizeof(float) +
      (size_t)(4 * 32) * sizeof(int);
  edge_pass_kernel<1><<<dim3(2048), dim3(128), edge_lds, stream>>>(
      pos, ei, pW1, pb1, pW2, pb2, aW1, ab1, aW2, ab2, a_src, a_dst, xv, amax,
      esum, outkey, E, N);
  edge_pass_kernel<2><<<dim3(2048), dim3(128), edge_lds, stream>>>(
      pos, ei, pW1, pb1, pW2, pb2, aW1, ab1, aW2, ab2, a_src, a_dst, xv, amax,
      esum, outkey, E, N);
  edge_pass_kernel<3><<<dim3(2048), dim3(128), edge_lds, stream>>>(
      pos, ei, pW1, pb1, pW2, pb2, aW1, ab1, aW2, ab2, a_src, a_dst, xv, amax,
      esum, outkey, E, N);

  const size_t out_lds = (size_t)(32 * P2) * sizeof(float);
  out_kernel<<<dim3(512), dim3(128), out_lds, stream>>>(outkey, W_out, b_out,
                                                        (float*)d_out, N);
}